// Edge_27547920236817
// MI455X (gfx1250) — compile-verified
//
#include <hip/hip_runtime.h>
#include <stdint.h>

// MI455X / gfx1250 edge-detector kernel.
// Memory-bound (453 MB min traffic @ 23.3 TB/s ~= 19.5 us floor). Inputs are
// staged once per tile via the CDNA5 async global->LDS DMA path (ASYNCcnt +
// s_wait_asynccnt), compute runs out of LDS, and the 403 MB output streams
// through non-temporal stores so L2 keeps the input working set.
// Transcendental pressure (the only thing that could exceed the memory floor)
// is minimized with gfx1250's hardware v_tanh_f32: sigmoid(10z)=0.5+0.5*tanh(5z)
// (1 trans op/sigmoid instead of exp+rcp), and sigmoid(-x)=1-sigmoid(x) folds
// the +/- direction pairs onto a single tanh.

#define IMG_H 512
#define IMG_W 512
#define NB    16
#define NC    3
#define TH    8
#define TW    32
#define RS    (TW + 2)          // 34 cols  (w-1 .. w+32)
#define TROWS (TH + 3)          // 11 rows  (h-2 .. h+8)
#define PLANE (TROWS * RS)      // 374
#define NE    (NC * PLANE)      // 1122 floats staged per block

__device__ __forceinline__ float gtf(float z) {
  // _gt(z) = sigmoid(10*z)
#if __has_builtin(__builtin_amdgcn_tanhf)
  return __builtin_fmaf(0.5f, __builtin_amdgcn_tanhf(5.0f * z), 0.5f);
#else
  float e = __builtin_amdgcn_exp2f(-14.4269504088896340736f * z);
  return __builtin_amdgcn_rcpf(1.0f + e);
#endif
}

// sigmoid(10z) and sigmoid(-10z) sharing one transcendental
__device__ __forceinline__ void gtf_pm(float z, float& sp, float& sn) {
#if __has_builtin(__builtin_amdgcn_tanhf)
  float th = 0.5f * __builtin_amdgcn_tanhf(5.0f * z);
  sp = 0.5f + th;
  sn = 0.5f - th;
#else
  sp = gtf(z);
  sn = 1.0f - sp;
#endif
}

// LDS tile accessor: channel ch, row offset dr in [-2,1], col offset dc in [-1,1]
#define SM(ch, dr, dc) smem[(ch) * PLANE + (lh + 2 + (dr)) * RS + (lw + 1 + (dc))]

__global__ __launch_bounds__(256) void edge_kernel(const float* __restrict__ x,
                                                   float* __restrict__ out) {
  __shared__ float smem[NE];
  const int tid = threadIdx.x;
  const int w0 = blockIdx.x * TW;
  const int h0 = blockIdx.y * TH;
  const int b  = blockIdx.z;

  // ---- Stage halo tile (all 3 channels) into LDS with async DMA ----------
  // global_load_async_to_lds writes LDS directly (no VGPR round-trip) and is
  // tracked by ASYNCcnt; low 32 bits of a flat shared pointer are the LDS
  // byte offset on gfx1250 (flat->LDS mapping truncates to addr[31:0]).
  const unsigned lds_base = (unsigned)(uintptr_t)&smem[0];
  for (int i = tid; i < NE; i += 256) {
    int c   = i / PLANE;
    int rem = i - c * PLANE;
    int r   = rem / RS;
    int col = rem - r * RS;
    int gh = h0 - 2 + r;   gh = gh < 0 ? 0 : (gh > IMG_H - 1 ? IMG_H - 1 : gh);
    int gw = w0 - 1 + col; gw = gw < 0 ? 0 : (gw > IMG_W - 1 ? IMG_W - 1 : gw);
    const float* gp = x + (size_t)(b * NC + c) * (IMG_H * IMG_W)
                        + (size_t)gh * IMG_W + gw;
    unsigned lo = lds_base + (unsigned)i * 4u;
    asm volatile("global_load_async_to_lds_b32 %0, %1, off"
                 :: "v"(lo), "v"(gp) : "memory");
  }
  asm volatile("s_wait_asynccnt 0x0" ::: "memory");
  __syncthreads();

  // ---- Per-thread output pixel -------------------------------------------
  const int lw = tid & 31;
  const int lh = tid >> 5;
  const int h = h0 + lh;
  const int w = w0 + lw;
  // Reference crops margin 2 then zero-pads back: border outputs are 0.
  const float msk = (h >= 2 && h < IMG_H - 2 && w >= 2 && w < IMG_W - 2) ? 1.0f : 0.0f;

  const size_t plane = (size_t)IMG_H * IMG_W;

#pragma unroll
  for (int c = 0; c < NC; ++c) {
    const int cm = (c + 2) % 3;   // rl(. , (+1,.)) channel source
    const int cp = (c + 1) % 3;   // rl(. , (-1,.)) channel source

    // Own-channel 3x3 neighborhood + (h-2, w)
    float x00   = SM(c,  0,  0);
    float pm1m1 = SM(c, -1, -1), pm10 = SM(c, -1, 0), pm1p1 = SM(c, -1, 1);
    float p0m1  = SM(c,  0, -1),                      p0p1  = SM(c,  0, 1);
    float pp1m1 = SM(c,  1, -1), pp10 = SM(c,  1, 0), pp1p1 = SM(c,  1, 1);
    float pm20  = SM(c, -2,  0);

    // Central differences (roll(s)[p] = im[p - s])
    float d11   = pm1m1 - x00;   // roll(+1,+1)
    float dnn11 = pp1p1 - x00;   // roll(-1,-1)
    float d01   = p0m1  - x00;   // roll(0,+1)
    float dn01  = p0p1  - x00;   // roll(0,-1)
    float d10   = pm10  - x00;   // roll(+1,0)
    float dn10  = pp10  - x00;   // roll(-1,0)
    float dn11  = pp1m1 - x00;   // roll(-1,+1)
    float d1n1  = pm1p1 - x00;   // roll(+1,-1)

    // Neighbor-channel samples for the rl(...) terms (axes = (C, H)!)
    float xm00   = SM(cm, 0,  0), xm0m1 = SM(cm, 0, -1), xm0p1 = SM(cm, 0, 1);
    float xmm10  = SM(cm, -1, 0), xmm2p1 = SM(cm, -2, 1), xmm2m1 = SM(cm, -2, -1);
    float xpm2m1 = SM(cp, -2, -1), xpm10 = SM(cp, -1, 0), xp0m1 = SM(cp, 0, -1);

    float m_d01   = xm0m1  - xm00;    // diff_01  @ (cm, h,   w)
    float m_dn01  = xm0p1  - xm00;    // diff_n01 @ (cm, h,   w)
    float m_dn11s = xm0m1  - xmm10;   // diff_n11 @ (cm, h-1, w)
    float m_d1n1s = xmm2p1 - xmm10;   // diff_1n1 @ (cm, h-1, w)
    float m_d11s  = xmm2m1 - xmm10;   // diff_11  @ (cm, h-1, w)
    float p_d11s  = xpm2m1 - xpm10;   // diff_11  @ (cp, h-1, w)
    float p_dn11s = xp0m1  - xpm10;   // diff_n11 @ (cp, h-1, w)
    float c_d10s  = pm20   - pm10;    // diff_10  @ (c,  h-1, w)
    float c_dn10s = x00    - pm10;    // diff_n10 @ (c,  h-1, w)
    float c_d01s  = pm1m1  - pm10;    // diff_01  @ (c,  h-1, w)

    // ---- group 10 ----
    float a10  = __builtin_fabsf(d10);
    float s0   = gtf(a10);                 // _gt(|d10| - DTR), DTR=0
    float ad10 = a10 * s0;
    float e10a = gtf(ad10 - __builtin_fabsf(d01))
               + gtf(ad10 - __builtin_fabsf(dn01))
               + gtf(ad10 - __builtin_fabsf(dn10));
    float t1   = gtf(ad10 - __builtin_fabsf(m_d01));     // used twice in ref
    float e10b = t1 + gtf(ad10 - __builtin_fabsf(m_dn01)) + t1;
    float g10  = gtf(e10a + e10b - 4.0f);
    float sp10, sn10;
    gtf_pm(d10, sp10, sn10);
    float e10  = g10 * sp10;
    float e10n = g10 * sn10;

    // ---- group 01 (gate reuses ad_10 per reference) ----
    float ad01 = __builtin_fabsf(d01) * gtf(ad10);
    float e01a = gtf(ad01 - __builtin_fabsf(d10))
               + gtf(ad01 - __builtin_fabsf(dn10))
               + gtf(ad01 - __builtin_fabsf(dn01));
    float e01b = gtf(ad01 - __builtin_fabsf(c_d10s))
               + gtf(ad01 - __builtin_fabsf(c_dn10s))
               + gtf(ad01 - __builtin_fabsf(c_d01s));
    float g01  = gtf(e01a + e01b - 4.0f);
    float e01  = g01 * gtf(d01);
    float e01n = e01;                       // reference uses +diff_01 for both

    // ---- group 11 ----
    float a11  = __builtin_fabsf(d11);
    float ad11 = a11 * gtf(a11);
    float e11a = gtf(ad11 - __builtin_fabsf(dn11))
               + gtf(ad11 - __builtin_fabsf(d1n1))
               + gtf(ad11 - __builtin_fabsf(dnn11));
    float e11b = gtf(ad11 - __builtin_fabsf(m_dn11s))
               + gtf(ad11 - __builtin_fabsf(m_d1n1s))
               + gtf(ad11 - __builtin_fabsf(m_d11s));
    float g11  = gtf(e11a + e11b - 4.0f);
    float sp11, sn11;
    gtf_pm(d11, sp11, sn11);
    float e11  = g11 * sp11;
    float e11n = g11 * sn11;

    // ---- group n11 (hard threshold at 0) ----
    float an  = __builtin_fabsf(dn11);
    float adn = (an > 0.0f) ? an : 0.0f;
    float ena = gtf(adn - __builtin_fabsf(d11))
              + gtf(adn - __builtin_fabsf(d1n1))
              + gtf(adn - __builtin_fabsf(dnn11));
    float t2  = gtf(adn - __builtin_fabsf(p_dn11s));     // used twice in ref
    float enb = gtf(adn - __builtin_fabsf(p_d11s)) + t2 + t2;
    float gn  = gtf(ena + enb - 4.0f);
    float spn, snn;
    gtf_pm(dn11, spn, snn);
    float en11  = gn * spn;
    float en11n = gn * snn;

    // ---- streaming stores: out[b, c*8 + k, h, w] -----------------------
    float* o = out + ((size_t)(b * (NC * 8) + c * 8)) * plane
                   + (size_t)h * IMG_W + w;
    __builtin_nontemporal_store(msk * e10,   o + 0 * plane);
    __builtin_nontemporal_store(msk * e10n,  o + 1 * plane);
    __builtin_nontemporal_store(msk * e01,   o + 2 * plane);
    __builtin_nontemporal_store(msk * e01n,  o + 3 * plane);
    __builtin_nontemporal_store(msk * e11,   o + 4 * plane);
    __builtin_nontemporal_store(msk * e11n,  o + 5 * plane);
    __builtin_nontemporal_store(msk * en11,  o + 6 * plane);
    __builtin_nontemporal_store(msk * en11n, o + 7 * plane);
  }
}

extern "C" void kernel_launch(void* const* d_in, const int* in_sizes, int n_in,
                              void* d_out, int out_size, void* d_ws, size_t ws_size,
                              hipStream_t stream) {
  (void)in_sizes; (void)n_in; (void)d_ws; (void)ws_size; (void)out_size;
  const float* x = (const float*)d_in[0];
  float* out = (float*)d_out;
  dim3 grid(IMG_W / TW, IMG_H / TH, NB);   // 16 x 64 x 16
  dim3 block(256);                          // 8 wave32
  edge_kernel<<<grid, block, 0, stream>>>(x, out);
}